// ModelNew_32607391711543
// MI455X (gfx1250) — compile-verified
//
#include <hip/hip_runtime.h>
#include <stdint.h>

typedef float v2f __attribute__((ext_vector_type(2)));
typedef float v8f __attribute__((ext_vector_type(8)));

#define CIN   16
#define COUT  32
#define SC    148104   // 34*66*66 (c_in stride in x)
#define SD    4356     // 66*66
#define SH    66
#define NWFRAG 13824   // 32*16*27 weight floats, stored in fragment order
#define XSTR  20       // padded cin-row stride (16 + 4 pad): bank-conflict free
#define NXS   (4*4*18*XSTR)   // 5760 floats staged x tile [d=4][h=4][w=18][cin]

// CDNA5 async gather: one 32-bit element, arbitrary per-lane global src ->
// arbitrary per-lane LDS dst (the transpose/swizzle happens inside the copy).
// Tracked by ASYNCcnt.
__device__ __forceinline__ void async_copy_b32(uint32_t lds_byte_addr, const void* gptr) {
    asm volatile("global_load_async_to_lds_b32 %0, %1, off"
                 :: "v"(lds_byte_addr), "v"((uint64_t)(uintptr_t)gptr)
                 : "memory");
}
__device__ __forceinline__ void wait_async0() {
    asm volatile("s_wait_asynccnt 0" ::: "memory");
}

// One workgroup = 128 threads = 4 waves = one 2x2x2 pool block x 16 W outputs.
// Wave (dd,hh) computes M=16 W-positions x N=32 channels; K = 16 c_in * 27 taps
// via v_wmma_f32_16x16x4_f32 (exact fp32 matrix path). Fully unrolled taps:
// every LDS operand is a ds_load with a compile-time immediate offset off one
// of two per-lane base addresses.
__global__ __launch_bounds__(128) void conv_pool_partial(
    const float* __restrict__ x, const float* __restrict__ wgt,
    float* __restrict__ partial)
{
    __shared__ float sWf[NWFRAG];   // weights in WMMA fragment order
    __shared__ float xs[NXS];       // x tile transposed: [d][h][w][cin(+pad)]
    __shared__ float smax[256];     // pooled maxes: 8 wpool * 32 channels
    __shared__ float ssum;

    const int tid  = threadIdx.x;
    const int bidx = blockIdx.x >> 11;       // 2048 blocks per batch
    const int rem  = blockIdx.x & 2047;
    const int dp   = rem >> 7;               // 16 pooled D
    const int hp   = (rem >> 2) & 31;        // 32 pooled H
    const int wt   = rem & 3;                // 4 tiles of 16 W
    const int w0   = wt * 16;

    const float* xb = x + (size_t)bidx * (CIN * SC);
    const size_t xoff0 = (size_t)(2*dp)*SD + (size_t)(2*hp)*SH + (size_t)w0;

    // ---- async stage weights, gathered into fragment order ----
    // fragment f = (tap*4 + c4)*2 + ntile; entry = f*64 + lane*2 + e
#pragma unroll 4
    for (int it = 0; it < NWFRAG / 128; ++it) {
        const int g    = tid + 128 * it;
        const int e    = g & 1;
        const int lane = (g >> 1) & 31;
        const int f    = g >> 6;
        const int t    = f & 1;
        const int c4   = (f >> 1) & 3;
        const int tap  = f >> 3;
        const int n    = (lane & 15) + 16 * t;
        const int cin  = 4 * c4 + 2 * (lane >> 4) + e;
        const int src  = (n * CIN + cin) * 27 + tap;
        async_copy_b32((uint32_t)(uintptr_t)&sWf[g], wgt + src);
    }
    // ---- async stage x tile, transposed so cin pairs are contiguous ----
    // element id = tid + 128*it over 4608 = 256 rows * 18 w; strength-reduced
    // row/w tracking (128 = 7*18 + 2), w innermost for coalesced global reads.
    {
        int wl = tid % 18;
        int rr = tid / 18;
        for (int it = 0; it < 36; ++it) {
            const int hh2 = rr & 3, dd2 = (rr >> 2) & 3, cin = rr >> 4;
            const float* src = xb + (size_t)cin * SC + xoff0
                             + (size_t)dd2 * SD + (size_t)hh2 * SH + wl;
            const uint32_t dst = (uint32_t)(uintptr_t)
                &xs[((dd2 * 4 + hh2) * 18 + wl) * XSTR + cin];
            async_copy_b32(dst, src);
            rr += 7; wl += 2;
            if (wl >= 18) { wl -= 18; rr += 1; }
        }
    }
    smax[tid]       = -3.402823466e38f;
    smax[tid + 128] = -3.402823466e38f;
    if (tid == 0) ssum = 0.f;
    wait_async0();
    __syncthreads();

    const int wave = tid >> 5, lane = tid & 31;
    const int dd = wave >> 1, hh = wave & 1;
    const int laneM = lane & 15, laneHi = lane >> 4;

    // Per-lane base addresses; all tap/chunk offsets below are constants.
    const float* abase = &xs[((dd * 4 + hh) * 18 + laneM) * XSTR + 2 * laneHi];
    const float* wbase = &sWf[lane * 2];

    v8f acc0 = {}; v8f acc1 = {};

#pragma unroll
    for (int kd = 0; kd < 3; ++kd)
#pragma unroll
    for (int kh = 0; kh < 3; ++kh)
#pragma unroll
    for (int kw = 0; kw < 3; ++kw) {
        const int tap  = (kd * 3 + kh) * 3 + kw;
        const int aoff = ((kd * 4 + kh) * 18 + kw) * XSTR;  // constant
#pragma unroll
        for (int c4 = 0; c4 < 4; ++c4) {
            // A (16x4 f32): v0 holds K=0/2 by lane half, v1 holds K=1/3
            const v2f a  = *(const v2f*)(abase + aoff + 4 * c4);
            const int f0 = (tap * 4 + c4) * 2;
            const v2f b0 = *(const v2f*)(wbase + (f0    ) * 64);
            const v2f b1 = *(const v2f*)(wbase + (f0 + 1) * 64);
            acc0 = __builtin_amdgcn_wmma_f32_16x16x4_f32(
                       false, a, false, b0, (short)0, acc0, false, false);
            acc1 = __builtin_amdgcn_wmma_f32_16x16x4_f32(
                       false, a, false, b1, (short)0, acc1, false, false);
        }
    }

    // Register W-pair max: D layout VGPR r -> M = r + 8*laneHi
#pragma unroll
    for (int j = 0; j < 4; ++j) {
        const float p0 = fmaxf(acc0[2*j], acc0[2*j+1]);
        const float p1 = fmaxf(acc1[2*j], acc1[2*j+1]);
        const int wpool = laneHi * 4 + j;
        atomicMax(&smax[wpool * 32 + laneM],      p0);  // cross-wave (dd,hh) max
        atomicMax(&smax[wpool * 32 + 16 + laneM], p1);
    }
    __syncthreads();
    const float v = smax[tid] + smax[tid + 128];
    atomicAdd(&ssum, v);
    __syncthreads();
    if (tid == 0) partial[blockIdx.x] = ssum;
}

// out[b] = 0.5 * S_b + sum_n(8192*conv_bias[n] + bias[n])
__global__ __launch_bounds__(256) void final_reduce(
    const float* __restrict__ partial, const float* __restrict__ conv_bias,
    const float* __restrict__ bias, float* __restrict__ out)
{
    __shared__ float red[256];
    const int b = blockIdx.x, t = threadIdx.x;
    float s = 0.f;
    for (int i = t; i < 2048; i += 256) s += partial[b * 2048 + i];
    red[t] = s;
    __syncthreads();
    for (int off = 128; off > 0; off >>= 1) {
        if (t < off) red[t] += red[t + off];
        __syncthreads();
    }
    if (t == 0) {
        float c = 0.f;
        for (int n = 0; n < COUT; ++n)
            c += 8192.0f * conv_bias[n] + bias[n];
        out[b] = 0.5f * red[0] + c;
    }
}

extern "C" void kernel_launch(void* const* d_in, const int* in_sizes, int n_in,
                              void* d_out, int out_size, void* d_ws, size_t ws_size,
                              hipStream_t stream) {
    const float* x    = (const float*)d_in[0];  // [8,16,34,66,66]
    const float* cw   = (const float*)d_in[1];  // [32,16,3,3,3]
    const float* cb   = (const float*)d_in[2];  // [32]
    const float* bias = (const float*)d_in[3];  // [32]
    float* out = (float*)d_out;                 // 8 floats
    float* ws  = (float*)d_ws;                  // 16384 partials (64 KB)

    conv_pool_partial<<<8 * 2048, 128, 0, stream>>>(x, cw, ws);
    final_reduce<<<8, 256, 0, stream>>>(ws, cb, bias, out);
}